// MoE_8383776161864
// MI455X (gfx1250) — compile-verified
//
#include <hip/hip_runtime.h>
#include <math.h>

// Problem constants (fixed by the reference).
#define NE 8        // experts
#define TOPK 2
#define H 1024      // hidden
#define II 512      // intermediate
#define T 4096      // tokens = B*S
#define TM 32       // token tile (two WMMA M-tiles) -- halves L2 weight traffic
#define MIDS (II + 2)   // LDS row stride for mid (bank-conflict pad)

typedef float v2f __attribute__((ext_vector_type(2)));
typedef float v8f __attribute__((ext_vector_type(8)));

// ---------------------------------------------------------------------------
// Kernel 0: zero the output and the per-expert counters.
// ---------------------------------------------------------------------------
__global__ __launch_bounds__(256) void moe_zero_kernel(float4* __restrict__ out4,
                                                       int n4,
                                                       int* __restrict__ counts) {
  int i = blockIdx.x * 256 + threadIdx.x;
  if (i < n4) out4[i] = make_float4(0.f, 0.f, 0.f, 0.f);
  if (blockIdx.x == 0 && threadIdx.x < NE) counts[threadIdx.x] = 0;
}

// ---------------------------------------------------------------------------
// Kernel 1: router. One thread per token: logits -> top-2 -> softmax -> append
// (token, gate) to the two selected experts' buckets.
// ---------------------------------------------------------------------------
__global__ __launch_bounds__(256) void moe_router_kernel(
    const float* __restrict__ x, const float* __restrict__ wg,
    int* __restrict__ counts, int* __restrict__ btok, float* __restrict__ bgate) {
  __shared__ float swg[NE * H];  // 32 KB
  for (int i = threadIdx.x; i < NE * H; i += 256) swg[i] = wg[i];
  __syncthreads();

  int t = blockIdx.x * 256 + threadIdx.x;
  if (t >= T) return;
  const float* xr = x + (size_t)t * H;

  float acc[NE];
#pragma unroll
  for (int e = 0; e < NE; ++e) acc[e] = 0.f;
  for (int h = 0; h < H; ++h) {
    float xv = xr[h];
#pragma unroll
    for (int e = 0; e < NE; ++e) acc[e] = fmaf(xv, swg[e * H + h], acc[e]);
  }

  // top-2 (ties -> lower index, like lax.top_k)
  int e0 = 0; float v0 = acc[0];
#pragma unroll
  for (int e = 1; e < NE; ++e) if (acc[e] > v0) { v0 = acc[e]; e0 = e; }
  int e1 = -1; float v1 = -3.4e38f;
#pragma unroll
  for (int e = 0; e < NE; ++e)
    if (e != e0 && acc[e] > v1) { v1 = acc[e]; e1 = e; }

  float g0 = 1.0f / (1.0f + expf(v1 - v0));  // softmax over {v0,v1}, stable
  float g1 = 1.0f - g0;

  int s0 = atomicAdd(&counts[e0], 1);
  btok[e0 * T + s0] = t;  bgate[e0 * T + s0] = g0;
  int s1 = atomicAdd(&counts[e1], 1);
  btok[e1 * T + s1] = t;  bgate[e1 * T + s1] = g1;
}

// ---------------------------------------------------------------------------
// Kernel 2: grouped expert FFN on a 32-token tile using f32 WMMA (16x16x4).
// Block = 256 threads = 8 waves. grid = (E, T/TM); blocks past count[e] exit.
// Stage 1: mid = gelu(X_tile @ Wfc[e]^T)    (M=32, K=H,  N=II; 64 cols/wave)
// Stage 2: out += gate * (mid @ Wproj[e]^T) (M=32, K=II, N=H; 128 cols/wave)
// Each B fragment is reused by the two M-tiles -> 8 wmma per 6 b64 loads.
// ---------------------------------------------------------------------------
__global__ __launch_bounds__(256) void moe_expert_kernel(
    const float* __restrict__ x, const float* __restrict__ wfc,
    const float* __restrict__ wproj, const int* __restrict__ counts,
    const int* __restrict__ btok, const float* __restrict__ bgate,
    float* __restrict__ out) {
  const int e = blockIdx.x;
  const int cnt = counts[e];
  const int tile0 = blockIdx.y * TM;
  if (tile0 >= cnt) return;

  __shared__ float s_mid[TM * MIDS];  // 32 x 514 f32 (~66 KB), padded rows
  __shared__ int   s_tok[TM];
  __shared__ float s_gate[TM];

  const int tid = threadIdx.x;
  if (tid < TM) {
    int idx = tile0 + tid;
    if (idx < cnt) {
      s_tok[tid]  = btok[e * T + idx];
      s_gate[tid] = bgate[e * T + idx];
    } else {  // pad with a valid token, zero gate (contributes exactly +0)
      s_tok[tid]  = btok[e * T + tile0];
      s_gate[tid] = 0.f;
    }
  }
  __syncthreads();

  const int wave = tid >> 5;          // 0..7
  const int lane = tid & 31;
  const int lm   = lane & 15;         // M (A) or N (B,C) index within 16
  const int lk   = (lane >> 4) << 1;  // K sub-offset: 0 or 2
  const int mbase = (lane >> 4) * 8;  // C/D: VGPR r holds row r + mbase

  // ---------------- stage 1: fc + gelu ----------------
  const float* xrow0 = x + (size_t)s_tok[lm] * H + lk;        // M-tile 0 rows
  const float* xrow1 = x + (size_t)s_tok[lm + 16] * H + lk;   // M-tile 1 rows
  const float* wfc_e = wfc + (size_t)e * II * H;

  v8f acc1[2][4] = {};
  for (int k0 = 0; k0 < H; k0 += 4) {
    v2f a0 = *(const v2f*)(xrow0 + k0);
    v2f a1 = *(const v2f*)(xrow1 + k0);
#pragma unroll
    for (int nt = 0; nt < 4; ++nt) {
      int n = wave * 64 + nt * 16 + lm;
      v2f b = *(const v2f*)(wfc_e + (size_t)n * H + k0 + lk);
      acc1[0][nt] = __builtin_amdgcn_wmma_f32_16x16x4_f32(
          false, a0, false, b, (short)0, acc1[0][nt], false, false);
      acc1[1][nt] = __builtin_amdgcn_wmma_f32_16x16x4_f32(
          false, a1, false, b, (short)0, acc1[1][nt], false, false);
    }
  }
#pragma unroll
  for (int mt = 0; mt < 2; ++mt) {
#pragma unroll
    for (int nt = 0; nt < 4; ++nt) {
      int n = wave * 64 + nt * 16 + lm;
#pragma unroll
      for (int r = 0; r < 8; ++r) {
        float v = acc1[mt][nt][r];
        v = 0.5f * v * (1.0f + erff(v * 0.70710678118654752f));  // exact gelu
        s_mid[(mt * 16 + mbase + r) * MIDS + n] = v;
      }
    }
  }
  __syncthreads();

  // ---------------- stage 2: proj + gated scatter-add ----------------
  const float* wp_e = wproj + (size_t)e * H * II;
  v8f acc2[2][8] = {};
  for (int k0 = 0; k0 < II; k0 += 4) {
    v2f a0 = *(const v2f*)(&s_mid[lm * MIDS + k0 + lk]);         // ds_load_b64
    v2f a1 = *(const v2f*)(&s_mid[(lm + 16) * MIDS + k0 + lk]);  // conflict-free
#pragma unroll
    for (int nt = 0; nt < 8; ++nt) {
      int n = wave * 128 + nt * 16 + lm;
      v2f b = *(const v2f*)(wp_e + (size_t)n * II + k0 + lk);
      acc2[0][nt] = __builtin_amdgcn_wmma_f32_16x16x4_f32(
          false, a0, false, b, (short)0, acc2[0][nt], false, false);
      acc2[1][nt] = __builtin_amdgcn_wmma_f32_16x16x4_f32(
          false, a1, false, b, (short)0, acc2[1][nt], false, false);
    }
  }
#pragma unroll
  for (int mt = 0; mt < 2; ++mt) {
#pragma unroll
    for (int nt = 0; nt < 8; ++nt) {
      int n = wave * 128 + nt * 16 + lm;
#pragma unroll
      for (int r = 0; r < 8; ++r) {
        int m = mt * 16 + mbase + r;
        float v = acc2[mt][nt][r] * s_gate[m];
        atomicAdd(out + (size_t)s_tok[m] * H + n, v);  // 2 nonzero adds/elt
      }
    }
  }
}

// ---------------------------------------------------------------------------
extern "C" void kernel_launch(void* const* d_in, const int* in_sizes, int n_in,
                              void* d_out, int out_size, void* d_ws, size_t ws_size,
                              hipStream_t stream) {
  (void)in_sizes; (void)n_in; (void)out_size; (void)ws_size;
  const float* x     = (const float*)d_in[0];  // [T, H]
  const float* wgate = (const float*)d_in[1];  // [E, H]
  const float* wfc   = (const float*)d_in[2];  // [E, I, H]
  const float* wproj = (const float*)d_in[3];  // [E, H, I]
  float* out = (float*)d_out;                  // [T, H]

  // workspace layout: counts | bucket tokens | bucket gates
  int*   counts = (int*)d_ws;
  int*   btok   = (int*)((char*)d_ws + 256);
  float* bgate  = (float*)((char*)d_ws + 256 + (size_t)NE * T * sizeof(int));

  const int n4 = (T * H) / 4;
  moe_zero_kernel<<<(n4 + 255) / 256, 256, 0, stream>>>((float4*)out, n4, counts);
  moe_router_kernel<<<T / 256, 256, 0, stream>>>(x, wgate, counts, btok, bgate);
  dim3 grid(NE, T / TM);
  moe_expert_kernel<<<grid, 256, 0, stream>>>(x, wfc, wproj, counts, btok, bgate, out);
}